// GAT_4844723109936
// MI455X (gfx1250) — compile-verified
//
#include <hip/hip_runtime.h>
#include <hip/hip_bf16.h>

typedef __attribute__((ext_vector_type(2))) float v2f;
typedef __attribute__((ext_vector_type(8))) float v8f;

#define LEAKY(z) ((z) > 0.0f ? (z) : 0.2f * (z))
#define EPSV 1e-16f
#define FH 64          // hidden width for both GAT layers
#define NCLS 10
#define NGRAPH 512

static inline int cdiv_i(int a, int b) { return (a + b - 1) / b; }

// ---------- monotone float<->u32 key for atomicMax-based segment max ----------
__device__ inline unsigned fkey(float f) {
    unsigned u = __float_as_uint(f);
    return (u >> 31) ? ~u : (u ^ 0x80000000u);
}
__device__ inline float finv(unsigned k) {
    unsigned u = (k >> 31) ? (k ^ 0x80000000u) : ~k;
    return __uint_as_float(u);
}
#define NEG_INF_KEY 0x007FFFFFu   // fkey(-inf)

// ---------- fills ----------
__global__ void k_fill_f32(float* p, float v, int n) {
    int i = blockIdx.x * blockDim.x + threadIdx.x;
    if (i < n) p[i] = v;
}
__global__ void k_fill_u32(unsigned* p, unsigned v, int n) {
    int i = blockIdx.x * blockDim.x + threadIdx.x;
    if (i < n) p[i] = v;
}

// ---------- H[nrows x 64] = X[nrows x K] @ W[K x 64] via V_WMMA_F32_16X16X4_F32 ----------
// One wave computes a 16x64 row-block (4 column tiles). W staged TRANSPOSED in LDS:
// Wt[n][k] with row stride S = K+4 so a lane's B pair {kk, kk+1} is one ds_load_b64
// into an adjacent (even-aligned) VGPR pair, and the 32-lane access pattern covers all
// 64 LDS banks exactly once (S = 4 mod 64; cross-half offset 2 words cannot alias).
// A/B/D striping per CDNA5 ISA 7.12.2:
//   A 16x4 f32: lanes 0-15 -> M=lane, {K=kk,kk+1}; lanes 16-31 -> M=lane-16, {K=kk+2,kk+3}
//   B 4x16 f32: mirrored with N in place of M
//   D 16x16 f32: VGPR r: lanes 0-15 -> (M=r, N=lane); lanes 16-31 -> (M=r+8, N=lane-16)
template <int K>
__global__ __launch_bounds__(256) void k_gemm_wmma(const float* __restrict__ X,
                                                   const float* __restrict__ Wg,
                                                   float* __restrict__ H,
                                                   int nrows) {
    constexpr int S = K + 4;                    // padded transposed row stride (words)
    __shared__ float Wt[FH * S];
    for (int i = threadIdx.x; i < K * FH; i += 256) {
        const int k = i >> 6, n = i & 63;       // Wg is [K x 64] row-major
        Wt[n * S + k] = Wg[i];
    }
    __syncthreads();

    const int lane = threadIdx.x & 31;
    const int wave = (blockIdx.x * blockDim.x + threadIdx.x) >> 5;
    const int row0 = wave * 16;
    if (row0 >= nrows) return;                  // wave-uniform exit (EXEC stays full)

    const int half = lane >> 4;                 // 0: K pair (0,1)   1: K pair (2,3)
    const int nn = lane & 15;
    // clamp instead of predicate: duplicated tail rows only feed D rows we never store
    const int mc = min(row0 + nn, nrows - 1);
    const float* xrow = X + (size_t)mc * K;

    v8f acc0 = {}, acc1 = {}, acc2 = {}, acc3 = {};
#pragma unroll
    for (int k = 0; k < K; k += 4) {
        const int kk = k + half * 2;
        const float2 av = *(const float2*)(xrow + kk);       // one global_load_b64
        v2f a; a.x = av.x; a.y = av.y;
        const float* wt = &Wt[nn * S + kk];
        const float2 b0 = *(const float2*)(wt);              // ds_load_b64, bank-conflict-free
        const float2 b1 = *(const float2*)(wt + 16 * S);
        const float2 b2 = *(const float2*)(wt + 32 * S);
        const float2 b3 = *(const float2*)(wt + 48 * S);
        v2f b;
        b.x = b0.x; b.y = b0.y;
        acc0 = __builtin_amdgcn_wmma_f32_16x16x4_f32(false, a, false, b, (short)0, acc0, false, false);
        b.x = b1.x; b.y = b1.y;
        acc1 = __builtin_amdgcn_wmma_f32_16x16x4_f32(false, a, false, b, (short)0, acc1, false, false);
        b.x = b2.x; b.y = b2.y;
        acc2 = __builtin_amdgcn_wmma_f32_16x16x4_f32(false, a, false, b, (short)0, acc2, false, false);
        b.x = b3.x; b.y = b3.y;
        acc3 = __builtin_amdgcn_wmma_f32_16x16x4_f32(false, a, false, b, (short)0, acc3, false, false);
    }

    const int mbase = row0 + half * 8;
#pragma unroll
    for (int r = 0; r < 8; ++r) {
        const int mm = mbase + r;
        if (mm < nrows) {
            float* hrow = H + (size_t)mm * FH + nn;
            hrow[0]  = acc0[r];
            hrow[16] = acc1[r];
            hrow[32] = acc2[r];
            hrow[48] = acc3[r];
        }
    }
}

// ---------- per-node attention logits: as = H . a_src ; ad = H . a_dst ----------
__global__ void k_alpha(const float* __restrict__ H, const float* __restrict__ a_s,
                        const float* __restrict__ a_d, float* __restrict__ as_,
                        float* __restrict__ ad_, int nrows) {
    int node = blockIdx.x * blockDim.x + threadIdx.x;
    if (node >= nrows) return;
    const float* h = H + (size_t)node * FH;
    float s = 0.0f, d = 0.0f;
#pragma unroll 8
    for (int f = 0; f < FH; ++f) {
        float v = h[f];
        s += v * a_s[f];
        d += v * a_d[f];
    }
    as_[node] = s;
    ad_[node] = d;
}

// ---------- edge pass 1: logit + segment max (keys), stash logit ----------
__global__ void k_edge_max(const int* __restrict__ src, const int* __restrict__ dst,
                           const float* __restrict__ as_, const float* __restrict__ ad_,
                           unsigned* __restrict__ mkey, float* __restrict__ logit,
                           int E, int N) {
    int e = blockIdx.x * blockDim.x + threadIdx.x;
    if (e >= E + N) return;
    int s, d;
    if (e < E) { s = src[e]; d = dst[e]; } else { s = d = e - E; }   // self loops
    float z = LEAKY(as_[s] + ad_[d]);
    logit[e] = z;
    atomicMax(&mkey[d], fkey(z));
}

// ---------- edge pass 2: e = exp(z - m[dst]); denom[dst] += e ----------
__global__ void k_edge_exp(const int* __restrict__ src, const int* __restrict__ dst,
                           const unsigned* __restrict__ mkey, float* __restrict__ ebuf,
                           float* __restrict__ denom, int E, int N) {
    int e = blockIdx.x * blockDim.x + threadIdx.x;
    if (e >= E + N) return;
    int d = (e < E) ? dst[e] : (e - E);
    float z = ebuf[e];
    float ex = __expf(z - finv(mkey[d]));
    ebuf[e] = ex;
    atomicAdd(&denom[d], ex);
}

// ---------- edge pass 3: normalize to attention coefficient ----------
__global__ void k_edge_coef(const int* __restrict__ dst, float* __restrict__ ebuf,
                            const float* __restrict__ denom, int E, int N) {
    int e = blockIdx.x * blockDim.x + threadIdx.x;
    if (e >= E + N) return;
    int d = (e < E) ? dst[e] : (e - E);
    ebuf[e] = ebuf[e] / (denom[d] + EPSV);
}

// ---------- edge pass 4: agg[dst] += coef * H[src], one thread per (edge, feature) ----------
__global__ void k_edge_agg(const int* __restrict__ src, const int* __restrict__ dst,
                           const float* __restrict__ coef, const float* __restrict__ H,
                           float* __restrict__ agg, int E, int N) {
    int gid = blockIdx.x * blockDim.x + threadIdx.x;
    int e = gid >> 6;           // 64 features per edge
    int f = gid & 63;
    if (e >= E + N) return;
    int s, d;
    if (e < E) { s = src[e]; d = dst[e]; } else { s = d = e - E; }
    float c = coef[e];
    atomicAdd(&agg[(size_t)d * FH + f], c * H[(size_t)s * FH + f]);
}

// ---------- bias (+ optional ReLU) ----------
__global__ void k_bias_act(float* __restrict__ agg, const float* __restrict__ bias,
                           int nrows, int do_relu) {
    int gid = blockIdx.x * blockDim.x + threadIdx.x;
    if (gid >= nrows * FH) return;
    float v = agg[gid] + bias[gid & 63];
    if (do_relu) v = fmaxf(v, 0.0f);
    agg[gid] = v;
}

// ---------- mean-pool accumulation ----------
__global__ void k_pool(const float* __restrict__ h, const int* __restrict__ batch,
                       float* __restrict__ sums, float* __restrict__ counts, int nrows) {
    int gid = blockIdx.x * blockDim.x + threadIdx.x;
    int node = gid >> 6, f = gid & 63;
    if (node >= nrows) return;
    int g = batch[node];
    atomicAdd(&sums[(size_t)g * FH + f], h[(size_t)node * FH + f]);
    if (f == 0) atomicAdd(&counts[g], 1.0f);
}

// ---------- classifier head: out[g, c] = (sums[g]/cnt) . Wlin[:, c] + blin[c] ----------
__global__ void k_head(const float* __restrict__ sums, const float* __restrict__ counts,
                       const float* __restrict__ Wlin, const float* __restrict__ blin,
                       float* __restrict__ out) {
    int gid = blockIdx.x * blockDim.x + threadIdx.x;
    if (gid >= NGRAPH * NCLS) return;
    int g = gid / NCLS, c = gid % NCLS;
    float inv = 1.0f / fmaxf(counts[g], 1.0f);
    float acc = 0.0f;
#pragma unroll 8
    for (int f = 0; f < FH; ++f)
        acc += sums[(size_t)g * FH + f] * Wlin[f * NCLS + c];
    out[gid] = acc * inv + blin[c];
}

extern "C" void kernel_launch(void* const* d_in, const int* in_sizes, int n_in,
                              void* d_out, int out_size, void* d_ws, size_t ws_size,
                              hipStream_t stream) {
    const float* x    = (const float*)d_in[0];
    const int*   edge = (const int*)d_in[1];        // [2, E] flat: row 0 = src, row 1 = dst
    const int*   batch= (const int*)d_in[2];
    const float* W1   = (const float*)d_in[3];
    const float* a_s1 = (const float*)d_in[4];
    const float* a_d1 = (const float*)d_in[5];
    const float* b1   = (const float*)d_in[6];
    const float* W2   = (const float*)d_in[7];
    const float* a_s2 = (const float*)d_in[8];
    const float* a_d2 = (const float*)d_in[9];
    const float* b2   = (const float*)d_in[10];
    const float* Wlin = (const float*)d_in[11];
    const float* blin = (const float*)d_in[12];
    float* out = (float*)d_out;

    const int N   = in_sizes[2];
    const int E   = in_sizes[1] / 2;
    const int FIN = in_sizes[0] / N;
    const int EP  = E + N;
    const int* src = edge;
    const int* dst = edge + E;

    // ---- workspace layout (floats) ----
    float* ws = (float*)d_ws;
    float*    h      = ws;                         // N * 64
    float*    agg    = h + (size_t)N * FH;         // N * 64
    float*    as_    = agg + (size_t)N * FH;       // N
    float*    ad_    = as_ + N;                    // N
    unsigned* mkey   = (unsigned*)(ad_ + N);       // N
    float*    denom  = (float*)mkey + N;           // N
    float*    ebuf   = denom + N;                  // EP
    float*    sums   = ebuf + EP;                  // 512 * 64
    float*    counts = sums + NGRAPH * FH;         // 512

    const int T = 256;
    const int gemm_blocks = cdiv_i(cdiv_i(N, 16) * 32, T);   // one wave per 16-row block

    for (int layer = 0; layer < 2; ++layer) {
        const float* Xin = (layer == 0) ? x : agg;
        const int K = (layer == 0) ? FIN : FH;
        const float* W  = (layer == 0) ? W1 : W2;
        const float* As = (layer == 0) ? a_s1 : a_s2;
        const float* Ad = (layer == 0) ? a_d1 : a_d2;
        const float* B  = (layer == 0) ? b1 : b2;

        // h = Xin @ W  (WMMA f32, W staged transposed in LDS)
        if (K == 128)
            k_gemm_wmma<128><<<gemm_blocks, T, 0, stream>>>(Xin, W, h, N);
        else
            k_gemm_wmma<64><<<gemm_blocks, T, 0, stream>>>(Xin, W, h, N);
        // per-node attention projections
        k_alpha<<<cdiv_i(N, T), T, 0, stream>>>(h, As, Ad, as_, ad_, N);
        // reset stats + aggregation target (agg no longer needed as input past gemm)
        k_fill_u32<<<cdiv_i(N, T), T, 0, stream>>>(mkey, NEG_INF_KEY, N);
        k_fill_f32<<<cdiv_i(N, T), T, 0, stream>>>(denom, 0.0f, N);
        k_fill_f32<<<cdiv_i(N * FH, T), T, 0, stream>>>(agg, 0.0f, N * FH);
        // segment softmax over edges (incl. self loops)
        k_edge_max <<<cdiv_i(EP, T), T, 0, stream>>>(src, dst, as_, ad_, mkey, ebuf, E, N);
        k_edge_exp <<<cdiv_i(EP, T), T, 0, stream>>>(src, dst, mkey, ebuf, denom, E, N);
        k_edge_coef<<<cdiv_i(EP, T), T, 0, stream>>>(dst, ebuf, denom, E, N);
        // weighted scatter-aggregate
        k_edge_agg<<<cdiv_i(EP * FH, T), T, 0, stream>>>(src, dst, ebuf, h, agg, E, N);
        // bias (+ReLU for layer 1)
        k_bias_act<<<cdiv_i(N * FH, T), T, 0, stream>>>(agg, B, N, layer == 0 ? 1 : 0);
    }

    // global mean pool + linear head
    k_fill_f32<<<cdiv_i(NGRAPH * FH, T), T, 0, stream>>>(sums, 0.0f, NGRAPH * FH);
    k_fill_f32<<<cdiv_i(NGRAPH, T), T, 0, stream>>>(counts, 0.0f, NGRAPH);
    k_pool<<<cdiv_i(N * FH, T), T, 0, stream>>>(agg, batch, sums, counts, N);
    k_head<<<cdiv_i(NGRAPH * NCLS, T), T, 0, stream>>>(sums, counts, Wlin, blin, out);
}